// user_encoder_76501957476795
// MI455X (gfx1250) — compile-verified
//
#include <hip/hip_runtime.h>
#include <math.h>

typedef float v2f __attribute__((ext_vector_type(2)));
typedef float v8f __attribute__((ext_vector_type(8)));

#define NNEWS 50
#define NE 20
#define NK 20
#define ND 100
#define NA 400
#define NQ 200
#define NH 20
#define DKH 20   // dk per head = A/H

static __device__ __forceinline__ v8f wmma_f32(v2f a, v2f b, v8f c) {
  // V_WMMA_F32_16X16X4_F32: D = A(16x4) * B(4x16) + C(16x16), all f32
  return __builtin_amdgcn_wmma_f32_16x16x4_f32(
      /*neg_a=*/false, a, /*neg_b=*/false, b,
      /*c_mod=*/(short)0, c, /*reuse_a=*/false, /*reuse_b=*/false);
}

// ---------------------------------------------------------------------------
// 1) KGAT: fused  scores[m] = relu(cat[m,:] @ w1 + b1) . w2 + b2
//    cat rows m = flattened (n,e,k), 20000 x 300, built on-the-fly into LDS.
//    One wave per 16-row M-tile; 25 N-tiles x 75 K-steps of f32 WMMA.
// ---------------------------------------------------------------------------
__global__ __launch_bounds__(32) void kgat_score_kernel(
    const float* __restrict__ ent, const float* __restrict__ nent,
    const float* __restrict__ nrel, const float* __restrict__ w1,
    const float* __restrict__ b1, const float* __restrict__ w2,
    const float* __restrict__ b2, float* __restrict__ scores) {
  __shared__ float tile[16 * 300];  // 19.2 KB cat tile
  const int m0 = blockIdx.x * 16;
  const int lane = threadIdx.x;
  const int half = lane >> 4;
  const int lcol = lane & 15;

  // Build cat tile in LDS: cat = [entity | neigh_entity | neigh_relation]
  for (int idx = lane; idx < 16 * 300; idx += 32) {
    const int r = idx / 300, c = idx % 300;
    const int row = m0 + r;
    const int n = row / (NE * NK);
    const int rem = row % (NE * NK);
    const int e = rem / NK, k = rem % NK;
    float v;
    if (c < ND)          v = ent[(n * NE + e) * ND + c];
    else if (c < 2 * ND) v = nent[((size_t)(n * NE + e) * NK + k) * ND + (c - ND)];
    else                 v = nrel[((size_t)(n * NE + e) * NK + k) * ND + (c - 2 * ND)];
    tile[idx] = v;
  }
  __syncthreads();

  v8f sacc = {};  // per-row partial of relu(h1)*w2, striped over lanes(N)
  for (int jt = 0; jt < NA / 16; ++jt) {
    const int col = jt * 16 + lcol;
    v8f acc = {};
    for (int k0 = 0; k0 < 300; k0 += 4) {
      const int kk = k0 + 2 * half;
      v2f a, b;
      a.x = tile[lcol * 300 + kk];
      a.y = tile[lcol * 300 + kk + 1];
      b.x = w1[kk * NA + col];
      b.y = w1[(kk + 1) * NA + col];
      acc = wmma_f32(a, b, acc);
    }
    const float bias = b1[col];
    const float wv = w2[col];
#pragma unroll
    for (int r = 0; r < 8; ++r) {
      float h = acc[r] + bias;
      h = fmaxf(h, 0.0f);
      sacc[r] += h * wv;
    }
  }
  const float b2v = b2[0];
#pragma unroll
  for (int r = 0; r < 8; ++r) {
    float s = sacc[r];
    s += __shfl_xor(s, 1, 32);
    s += __shfl_xor(s, 2, 32);
    s += __shfl_xor(s, 4, 32);
    s += __shfl_xor(s, 8, 32);
    if (lane == 0)  scores[m0 + r] = s + b2v;        // M = r       (half 0)
    if (lane == 16) scores[m0 + 8 + r] = s + b2v;    // M = r + 8   (half 1)
  }
}

// ---------------------------------------------------------------------------
// 2) Softmax over K neighbors + weighted aggregation -> agg [N, E*2D]
// ---------------------------------------------------------------------------
__global__ __launch_bounds__(128) void kgat_agg_kernel(
    const float* __restrict__ ent, const float* __restrict__ nent,
    const float* __restrict__ scores, float* __restrict__ agg) {
  const int g = blockIdx.x;  // n*NE + e
  const int n = g / NE, e = g % NE;
  const int tid = threadIdx.x;
  __shared__ float w[NK];
  if (tid < NK) w[tid] = scores[g * NK + tid];
  __syncthreads();
  if (tid == 0) {
    float m = -3.4e38f;
    for (int k = 0; k < NK; ++k) m = fmaxf(m, w[k]);
    float s = 0.f;
    for (int k = 0; k < NK; ++k) { w[k] = expf(w[k] - m); s += w[k]; }
    const float inv = 1.f / s;
    for (int k = 0; k < NK; ++k) w[k] *= inv;
  }
  __syncthreads();
  float* aggrow = agg + (size_t)n * (NE * 2 * ND) + e * (2 * ND);
  for (int d = tid; d < ND; d += 128) {
    aggrow[d] = ent[g * ND + d];
    float acc = 0.f;
    const float* np = nent + (size_t)g * NK * ND + d;
    for (int k = 0; k < NK; ++k) acc += w[k] * np[k * ND];
    aggrow[ND + d] = acc;
  }
}

// ---------------------------------------------------------------------------
// 3) Generic f32 WMMA GEMM: C[M,N] = act(A[M,K] @ B[K,N] + bias)
//    One wave per block; each block owns one 16-col N-tile and up to 4
//    M-tiles (shares each B fragment across them -> B streamed once).
//    Requires K % 4 == 0. act: 0 = none, 1 = tanh.
// ---------------------------------------------------------------------------
__global__ __launch_bounds__(32) void wmma_gemm_kernel(
    const float* __restrict__ A, const float* __restrict__ B,
    const float* __restrict__ bias, float* __restrict__ C,
    int M, int K, int N, int act) {
  const int j0 = blockIdx.x * 16;
  const int mbase = blockIdx.y * 64;
  const int lane = threadIdx.x;
  const int half = lane >> 4;
  const int lcol = lane & 15;
  const int col = j0 + lcol;
  const bool colok = col < N;

  v8f acc[4] = {};
  for (int k0 = 0; k0 < K; k0 += 4) {
    const int kk = k0 + 2 * half;
    v2f b;
    b.x = colok ? B[(size_t)kk * N + col] : 0.f;
    b.y = colok ? B[(size_t)(kk + 1) * N + col] : 0.f;
#pragma unroll
    for (int mi = 0; mi < 4; ++mi) {
      const int row = mbase + mi * 16 + lcol;
      v2f a;
      a.x = (row < M) ? A[(size_t)row * K + kk] : 0.f;
      a.y = (row < M) ? A[(size_t)row * K + kk + 1] : 0.f;
      acc[mi] = wmma_f32(a, b, acc[mi]);
    }
  }
  const float bv = (bias != nullptr && colok) ? bias[col] : 0.f;
#pragma unroll
  for (int mi = 0; mi < 4; ++mi) {
#pragma unroll
    for (int r = 0; r < 8; ++r) {
      const int row = mbase + mi * 16 + 8 * half + r;
      if (row < M && colok) {
        float v = acc[mi][r] + bv;
        if (act == 1) v = tanhf(v);
        C[(size_t)row * N + col] = v;
      }
    }
  }
}

// ---------------------------------------------------------------------------
// 4) LayerNorm over last dim `width`, one block (256 thr) per row.
// ---------------------------------------------------------------------------
__global__ __launch_bounds__(256) void ln_kernel(
    const float* __restrict__ in, float* __restrict__ out,
    const float* __restrict__ g, const float* __restrict__ b, int width) {
  const int row = blockIdx.x;
  const int tid = threadIdx.x;
  const float* x = in + (size_t)row * width;
  __shared__ float red[256];
  float s = 0.f;
  for (int i = tid; i < width; i += 256) s += x[i];
  red[tid] = s;
  __syncthreads();
  for (int off = 128; off > 0; off >>= 1) {
    if (tid < off) red[tid] += red[tid + off];
    __syncthreads();
  }
  const float mean = red[0] / width;
  __syncthreads();
  float v = 0.f;
  for (int i = tid; i < width; i += 256) { float d = x[i] - mean; v += d * d; }
  red[tid] = v;
  __syncthreads();
  for (int off = 128; off > 0; off >>= 1) {
    if (tid < off) red[tid] += red[tid + off];
    __syncthreads();
  }
  const float inv = rsqrtf(red[0] / width + 1e-5f);
  for (int i = tid; i < width; i += 256)
    out[(size_t)row * width + i] = (x[i] - mean) * inv * g[i] + b[i];
}

// ---------------------------------------------------------------------------
// 5) Additive-attention pools (entity per news, user over news)
// ---------------------------------------------------------------------------
__global__ __launch_bounds__(256) void att_pool_entity_kernel(
    const float* __restrict__ t_e, const float* __restrict__ qv,
    const float* __restrict__ xln, float* __restrict__ out) {
  const int n = blockIdx.x;
  const int tid = threadIdx.x;
  __shared__ float w[NE];
  if (tid < NE) {
    float acc = 0.f;
    const float* tp = t_e + (size_t)(n * NE + tid) * NQ;
    for (int q = 0; q < NQ; ++q) acc += tp[q] * qv[q];
    w[tid] = acc;
  }
  __syncthreads();
  if (tid == 0) {
    float m = -3.4e38f;
    for (int e = 0; e < NE; ++e) m = fmaxf(m, w[e]);
    float s = 0.f;
    for (int e = 0; e < NE; ++e) { w[e] = expf(w[e] - m); s += w[e]; }
    const float inv = 1.f / s;
    for (int e = 0; e < NE; ++e) w[e] *= inv;
  }
  __syncthreads();
  for (int a = tid; a < NA; a += 256) {
    float acc = 0.f;
    for (int e = 0; e < NE; ++e) acc += w[e] * xln[(size_t)n * (NE * NA) + e * NA + a];
    out[n * NA + a] = acc;
  }
}

__global__ __launch_bounds__(256) void att_pool_user_kernel(
    const float* __restrict__ t_u, const float* __restrict__ qv,
    const float* __restrict__ oln, float* __restrict__ out) {
  const int tid = threadIdx.x;
  __shared__ float w[NNEWS];
  if (tid < NNEWS) {
    float acc = 0.f;
    const float* tp = t_u + tid * NQ;
    for (int q = 0; q < NQ; ++q) acc += tp[q] * qv[q];
    w[tid] = acc;
  }
  __syncthreads();
  if (tid == 0) {
    float m = -3.4e38f;
    for (int i = 0; i < NNEWS; ++i) m = fmaxf(m, w[i]);
    float s = 0.f;
    for (int i = 0; i < NNEWS; ++i) { w[i] = expf(w[i] - m); s += w[i]; }
    const float inv = 1.f / s;
    for (int i = 0; i < NNEWS; ++i) w[i] *= inv;
  }
  __syncthreads();
  for (int a = tid; a < NA; a += 256) {
    float acc = 0.f;
    for (int i = 0; i < NNEWS; ++i) acc += w[i] * oln[i * NA + a];
    out[a] = acc;
  }
}

// ---------------------------------------------------------------------------
// 6) Per-head attention over the news axis. One block per head; scores in LDS.
// ---------------------------------------------------------------------------
__global__ __launch_bounds__(256) void mhsa_kernel(
    const float* __restrict__ qb, const float* __restrict__ kb,
    const float* __restrict__ vb, float* __restrict__ o) {
  const int h = blockIdx.x;
  const int tid = threadIdx.x;
  __shared__ float sc[NNEWS * NNEWS];  // 10 KB
  const float scale = rsqrtf((float)DKH);
  for (int idx = tid; idx < NNEWS * NNEWS; idx += 256) {
    const int i = idx / NNEWS, j = idx % NNEWS;
    float dot = 0.f;
    for (int d = 0; d < DKH; ++d)
      dot += qb[i * NA + h * DKH + d] * kb[j * NA + h * DKH + d];
    sc[idx] = dot * scale;
  }
  __syncthreads();
  if (tid < NNEWS) {
    float m = -3.4e38f;
    for (int j = 0; j < NNEWS; ++j) m = fmaxf(m, sc[tid * NNEWS + j]);
    float s = 0.f;
    for (int j = 0; j < NNEWS; ++j) {
      const float ev = expf(sc[tid * NNEWS + j] - m);
      sc[tid * NNEWS + j] = ev;
      s += ev;
    }
    const float inv = 1.f / s;
    for (int j = 0; j < NNEWS; ++j) sc[tid * NNEWS + j] *= inv;
  }
  __syncthreads();
  for (int idx = tid; idx < NNEWS * DKH; idx += 256) {
    const int i = idx / DKH, d = idx % DKH;
    float acc = 0.f;
    for (int j = 0; j < NNEWS; ++j) acc += sc[i * NNEWS + j] * vb[j * NA + h * DKH + d];
    o[i * NA + h * DKH + d] = acc;
  }
}

// ---------------------------------------------------------------------------
extern "C" void kernel_launch(void* const* d_in, const int* in_sizes, int n_in,
                              void* d_out, int out_size, void* d_ws, size_t ws_size,
                              hipStream_t stream) {
  (void)in_sizes; (void)n_in; (void)out_size; (void)ws_size;
  const float* ent  = (const float*)d_in[0];
  const float* nent = (const float*)d_in[1];
  const float* nrel = (const float*)d_in[2];
  const float* kw1  = (const float*)d_in[3];
  const float* kb1  = (const float*)d_in[4];
  const float* kw2  = (const float*)d_in[5];
  const float* kb2  = (const float*)d_in[6];
  const float* gcnw = (const float*)d_in[7];
  const float* gcnb = (const float*)d_in[8];
  const float* ln3g = (const float*)d_in[9];
  const float* ln3b = (const float*)d_in[10];
  const float* aew  = (const float*)d_in[11];
  const float* aeb  = (const float*)d_in[12];
  const float* aeq  = (const float*)d_in[13];
  const float* ln4g = (const float*)d_in[14];
  const float* ln4b = (const float*)d_in[15];
  const float* wq   = (const float*)d_in[16];
  const float* bq   = (const float*)d_in[17];
  const float* wk   = (const float*)d_in[18];
  const float* bk   = (const float*)d_in[19];
  const float* wv   = (const float*)d_in[20];
  const float* bv   = (const float*)d_in[21];
  const float* ln1g = (const float*)d_in[22];
  const float* ln1b = (const float*)d_in[23];
  const float* auw  = (const float*)d_in[24];
  const float* aub  = (const float*)d_in[25];
  const float* auq  = (const float*)d_in[26];
  const float* ln2g = (const float*)d_in[27];
  const float* ln2b = (const float*)d_in[28];

  float* ws     = (float*)d_ws;
  float* scores = ws;                  // 20000
  float* agg    = scores + 20000;      // 50*4000   = 200000
  float* xbuf   = agg    + 200000;     // 50*8000   = 400000
  float* xln    = xbuf   + 400000;     // 400000
  float* t_e    = xln    + 400000;     // 1000*200  = 200000
  float* nr_raw = t_e    + 200000;     // 50*400    = 20000
  float* nrep   = nr_raw + 20000;      // 20000
  float* qb_    = nrep   + 20000;      // 20000
  float* kb_    = qb_    + 20000;      // 20000
  float* vb_    = kb_    + 20000;      // 20000
  float* ob     = vb_    + 20000;      // 20000
  float* obln   = ob     + 20000;      // 20000
  float* t_u    = obln   + 20000;      // 50*200    = 10000
  float* uraw   = t_u    + 10000;      // 400

  // 1) KGAT fused scores (WMMA, 4.8 GFLOP)
  kgat_score_kernel<<<(NNEWS * NE * NK) / 16, 32, 0, stream>>>(
      ent, nent, nrel, kw1, kb1, kw2, kb2, scores);
  // 2) neighbor softmax + aggregation -> agg [50,4000]
  kgat_agg_kernel<<<NNEWS * NE, 128, 0, stream>>>(ent, nent, scores, agg);
  // 3) GCN: xbuf = tanh(agg @ gcn_w + gcn_b)  [50,8000] (WMMA; streams 128MB once)
  wmma_gemm_kernel<<<dim3(8000 / 16, 1), 32, 0, stream>>>(
      agg, gcnw, gcnb, xbuf, NNEWS, 4000, 8000, 1);
  // 4) ln3 on 1000 rows of 400
  ln_kernel<<<NNEWS * NE, 256, 0, stream>>>(xbuf, xln, ln3g, ln3b, NA);
  // 5) t_e = tanh(xln[1000,400] @ att_e_w[400,200] + b)   (WMMA)
  wmma_gemm_kernel<<<dim3(13, 16), 32, 0, stream>>>(
      xln, aew, aeb, t_e, NNEWS * NE, NA, NQ, 1);
  // 6) additive pool over entities -> nr_raw [50,400]
  att_pool_entity_kernel<<<NNEWS, 256, 0, stream>>>(t_e, aeq, xln, nr_raw);
  // 7) ln4 -> new_rep
  ln_kernel<<<NNEWS, 256, 0, stream>>>(nr_raw, nrep, ln4g, ln4b, NA);
  // 8) Q/K/V projections (WMMA)
  wmma_gemm_kernel<<<dim3(NA / 16, 1), 32, 0, stream>>>(nrep, wq, bq, qb_, NNEWS, NA, NA, 0);
  wmma_gemm_kernel<<<dim3(NA / 16, 1), 32, 0, stream>>>(nrep, wk, bk, kb_, NNEWS, NA, NA, 0);
  wmma_gemm_kernel<<<dim3(NA / 16, 1), 32, 0, stream>>>(nrep, wv, bv, vb_, NNEWS, NA, NA, 0);
  // 9) per-head attention
  mhsa_kernel<<<NH, 256, 0, stream>>>(qb_, kb_, vb_, ob);
  // 10) ln1
  ln_kernel<<<NNEWS, 256, 0, stream>>>(ob, obln, ln1g, ln1b, NA);
  // 11) t_u = tanh(obln @ att_u_w + b)   (WMMA)
  wmma_gemm_kernel<<<dim3(13, 1), 32, 0, stream>>>(obln, auw, aub, t_u, NNEWS, NA, NQ, 1);
  // 12) additive pool over news -> uraw [400]
  att_pool_user_kernel<<<1, 256, 0, stream>>>(t_u, auq, obln, uraw);
  // 13) ln2 -> d_out [1,400]
  ln_kernel<<<1, 256, 0, stream>>>(uraw, (float*)d_out, ln2g, ln2b, NA);
}